// PairwiseDistModule_88192858456731
// MI455X (gfx1250) — compile-verified
//
#include <hip/hip_runtime.h>

typedef float v2f __attribute__((ext_vector_type(2)));
typedef float v8f __attribute__((ext_vector_type(8)));

#define D_DIM    2560
#define NSUP     25
#define QSIZE    4096
#define NWAY     5
#define KSHOT    5
#define EPS      1e-6f
#define D_EPS2   (2560.0f * 1e-12f)   // D * eps^2

// ---------------------------------------------------------------------------
// Kernel 1: per-support-row term  sterm[n] = ||s_n||^2 + 2*eps*sum(s_n)
// ---------------------------------------------------------------------------
__global__ __launch_bounds__(128) void sterm_kernel(const float* __restrict__ support,
                                                    float* __restrict__ sterm) {
    __shared__ float red_sq[128];
    __shared__ float red_sm[128];
    const int n = blockIdx.x;               // 0..24
    const int t = threadIdx.x;
    const float* row = support + (size_t)n * D_DIM;
    float sq = 0.0f, sm = 0.0f;
    for (int k = t; k < D_DIM; k += 128) {
        float v = row[k];
        sq = fmaf(v, v, sq);
        sm += v;
    }
    red_sq[t] = sq; red_sm[t] = sm;
    __syncthreads();
    for (int s = 64; s > 0; s >>= 1) {
        if (t < s) { red_sq[t] += red_sq[t + s]; red_sm[t] += red_sm[t + s]; }
        __syncthreads();
    }
    if (t == 0) sterm[n] = red_sq[0] + 2.0f * EPS * red_sm[0];
}

// ---------------------------------------------------------------------------
// Kernel 2: 16-query-row tile per block; 4 waves split K; fp32 WMMA GEMM
//   dot[m][n] = sum_k query[m0+m][k] * support[n][k]
//   dist^2    = ||q||^2 + sterm[n] - 2*dot - 2*eps*sum(q) + D*eps^2
//   score[m][w] = -sum_{j<5} dist[m][5w+j]
// ---------------------------------------------------------------------------
__global__ __launch_bounds__(128) void pairwise_wmma_kernel(const float* __restrict__ query,
                                                            const float* __restrict__ support,
                                                            const float* __restrict__ sterm,
                                                            float* __restrict__ out) {
    __shared__ float c_lds[4][2][8][32];   // [wave][ntile][reg][lane]
    __shared__ float sq_lds[4][32];        // per-wave per-lane |q|^2 partials
    __shared__ float sm_lds[4][32];        // per-wave per-lane sum(q) partials
    __shared__ float qn_lds[16];
    __shared__ float qs_lds[16];
    __shared__ float st_lds[32];
    __shared__ float dist_lds[16][32];

    const int tid  = threadIdx.x;          // 0..127
    const int lane = tid & 31;             // wave32 lane
    const int wv   = tid >> 5;             // 0..3
    const int half = lane >> 4;            // 0 or 1  (K sub-pair in A/B frags)
    const int mrow = lane & 15;            // M row (A) / N col (B,C)
    const int m0   = blockIdx.x * 16;      // query tile base row

    if (tid < 32) st_lds[tid] = (tid < NSUP) ? sterm[tid] : 0.0f;

    // ---- phase 1: WMMA accumulate over this wave's K chunk --------------
    const int kbase = wv * 640;            // 2560/4 waves
    const int n0 = mrow;                   // tile0 support row
    const int n1 = (mrow + 16 < NSUP) ? (mrow + 16) : 0;   // tile1 (clamped pad)

    const float* qrow  = query   + (size_t)(m0 + mrow) * D_DIM + 2 * half;
    const float* srow0 = support + (size_t)n0 * D_DIM + 2 * half;
    const float* srow1 = support + (size_t)n1 * D_DIM + 2 * half;

    v8f c0 = {}; v8f c1 = {};
    float psq = 0.0f, psm = 0.0f;

    for (int k = kbase; k < kbase + 640; k += 4) {
        v2f a  = *(const v2f*)(qrow  + k);
        v2f b0 = *(const v2f*)(srow0 + k);
        v2f b1 = *(const v2f*)(srow1 + k);
        // D = A(16x4 f32) x B(4x16 f32) + C : v_wmma_f32_16x16x4_f32
        c0 = __builtin_amdgcn_wmma_f32_16x16x4_f32(false, a, false, b0,
                                                   (short)0, c0, false, false);
        c1 = __builtin_amdgcn_wmma_f32_16x16x4_f32(false, a, false, b1,
                                                   (short)0, c1, false, false);
        psq = fmaf(a.x, a.x, psq);
        psq = fmaf(a.y, a.y, psq);
        psm += a.x + a.y;
    }

    #pragma unroll
    for (int r = 0; r < 8; ++r) {
        c_lds[wv][0][r][lane] = c0[r];
        c_lds[wv][1][r][lane] = c1[r];
    }
    sq_lds[wv][lane] = psq;
    sm_lds[wv][lane] = psm;
    __syncthreads();

    // ---- phase 2: reduce |q|^2 and sum(q) per M row ---------------------
    if (tid < 16) {
        float qn = 0.0f, qs = 0.0f;
        #pragma unroll
        for (int w = 0; w < 4; ++w) {
            qn += sq_lds[w][tid] + sq_lds[w][tid + 16];
            qs += sm_lds[w][tid] + sm_lds[w][tid + 16];
        }
        qn_lds[tid] = qn;
        qs_lds[tid] = qs;
    }
    __syncthreads();

    // ---- phase 3: wave 0 reduces C across waves, forms distances --------
    // C layout (16x16 f32): VGPR r, lane l -> M = r + 8*(l/16), N = l%16
    if (tid < 32) {
        const int n = mrow;
        #pragma unroll
        for (int r = 0; r < 8; ++r) {
            const int m = r + 8 * half;
            float cs0 = 0.0f, cs1 = 0.0f;
            #pragma unroll
            for (int w = 0; w < 4; ++w) {
                cs0 += c_lds[w][0][r][tid];
                cs1 += c_lds[w][1][r][tid];
            }
            const float base = qn_lds[m] - 2.0f * EPS * qs_lds[m] + D_EPS2;
            dist_lds[m][n] = sqrtf(fmaxf(base + st_lds[n] - 2.0f * cs0, 0.0f));
            if (n + 16 < NSUP)
                dist_lds[m][n + 16] =
                    sqrtf(fmaxf(base + st_lds[n + 16] - 2.0f * cs1, 0.0f));
        }
    }
    __syncthreads();

    // ---- phase 4: scores -------------------------------------------------
    if (tid < 16 * NWAY) {
        const int m = tid / NWAY;
        const int w = tid % NWAY;
        float s = 0.0f;
        #pragma unroll
        for (int j = 0; j < KSHOT; ++j) s += dist_lds[m][w * KSHOT + j];
        out[(size_t)(m0 + m) * NWAY + w] = -s;
    }
}

// ---------------------------------------------------------------------------
extern "C" void kernel_launch(void* const* d_in, const int* in_sizes, int n_in,
                              void* d_out, int out_size, void* d_ws, size_t ws_size,
                              hipStream_t stream) {
    const float* support = (const float*)d_in[0];   // [25, 2560]
    const float* query   = (const float*)d_in[1];   // [4096, 2560]
    float* out   = (float*)d_out;                   // [4096, 5]
    float* sterm = (float*)d_ws;                    // 25 floats of scratch

    sterm_kernel<<<NSUP, 128, 0, stream>>>(support, sterm);
    pairwise_wmma_kernel<<<QSIZE / 16, 128, 0, stream>>>(query, support, sterm, out);
}